// TargetAwareAttention_12618613916204
// MI455X (gfx1250) — compile-verified
//
#include <hip/hip_runtime.h>

#define B_SZ   2048
#define EMB    512
#define L_MAX  128
#define NH     8
#define HD     64
#define LN_EPS 1e-5f

typedef __attribute__((ext_vector_type(2))) float    v2f;
typedef __attribute__((ext_vector_type(8))) float    v8f;
typedef __attribute__((ext_vector_type(4))) unsigned v4u;
typedef __attribute__((ext_vector_type(8))) int      v8i;
typedef __attribute__((ext_vector_type(4))) int      v4i;

// ---- LDS layout (float offsets) -------------------------------------------
#define BUF_STRIDE  68                 // 64 data + 4 pad (TDM pad_amount=4 DW)
#define QK_STRIDE   520                // 512 + 8 pad
#define W_STRIDE    132                // 128 + 4 pad
#define CTX_STRIDE  520
#define OFF_BUF0    0                  // 128 x 68 = 8704   (double buffer A)
#define OFF_BUF1    8704               // 128 x 68 = 8704   (double buffer B)
#define OFF_QK      17408              // 16 x 520 = 8320   (rows 8..15 zero)
#define OFF_W       25728              // 16 x 132 = 2112   (rows 8..15 zero)
#define OFF_CTX     27840              // 8 x 520  = 4160   (K-half 0 partial)
#define OFF_CTX2    32000              // 8 x 520  = 4160   (K-half 1 partial)
#define OFF_X       36160              // 512
#define OFF_Q       36672              // 512
#define OFF_AO      37184              // 512
#define OFF_QBK     37696              // 8
#define OFF_RED     37704              // 16
#define SMEM_FLOATS 37720              // ~147 KB -> 2 blocks / 320KB WGP

// ---- Tensor Data Mover: DMA one 128x64 f32 tile (row stride 512) to LDS ---
// D# group0: count=1 | lds_addr | global_addr(57b) | type=2
// D# group1: data_size=4B, pad_enable, pad_interval=64 DW, pad_amount=4 DW,
//            tensor_dim0=64, tensor_dim1=128, tile_dim0=64, tile_dim1=128,
//            tensor_dim0_stride=512
__device__ __forceinline__ void tdm_load_tile64(const float* gsrc, float* ldst)
{
    unsigned long long ga = (unsigned long long)gsrc;
    unsigned lds_off = (unsigned)(unsigned long long)ldst;  // low 32b = LDS byte offset
    v4u g0;
    g0[0] = 1u;                                   // count = 1 (valid user D#)
    g0[1] = lds_off;                              // lds_addr
    g0[2] = (unsigned)ga;                         // global_addr[31:0]
    g0[3] = (unsigned)((ga >> 32) & 0x01ffffffu)  // global_addr[56:32]
          | (2u << 30);                           // type = 2 ("image")
    v8i g1;
    g1[0] = (int)((2u << 16)                      // data_size = 4 bytes
                | (1u << 20)                      // pad_enable
                | (5u << 22)                      // pad_interval: 64 DWORDs
                | (3u << 25));                    // pad_amount : 4 DWORDs
    g1[1] = (int)(64u  << 16);                    // tensor_dim0 = 64
    g1[2] = (int)(128u << 16);                    // tensor_dim1 = 128
    g1[3] = (int)(64u  << 16);                    // tile_dim0   = 64
    g1[4] = 128;                                  // tile_dim1   = 128
    g1[5] = 512;                                  // tensor_dim0_stride = 512
    g1[6] = 0;
    g1[7] = 0;
    v4i gz4 = {0, 0, 0, 0};                       // 2D tile: groups 2/3 unused
    v8i gz8 = {0, 0, 0, 0, 0, 0, 0, 0};
    // clang-23 / therock-10.0 toolchain: 6-arg form
    __builtin_amdgcn_tensor_load_to_lds(g0, g1, gz4, gz4, gz8, 0);
}

__global__ void TargetAwareAttention_fused(
    const float* __restrict__ x_g,
    const float* __restrict__ keys_g,
    const float* __restrict__ vals_g,
    const int*   __restrict__ len_g,
    const float* __restrict__ Wq, const float* __restrict__ bq,
    const float* __restrict__ Wk, const float* __restrict__ bk,
    const float* __restrict__ Wv, const float* __restrict__ bv,
    const float* __restrict__ Wo, const float* __restrict__ bo,
    const float* __restrict__ gamma, const float* __restrict__ beta,
    float* __restrict__ out_norm,
    float* __restrict__ out_w)
{
    extern __shared__ float smem[];
    const int b    = blockIdx.x;
    const int t    = threadIdx.x;          // 256 threads = 8 waves (wave32)
    const int wid  = t >> 5;
    const int lane = t & 31;
    const int half = lane >> 4;
    const int lm   = lane & 15;
    const int len  = len_g[b];

    float* s_buf0 = smem + OFF_BUF0;
    float* s_buf1 = smem + OFF_BUF1;
    float* s_qk   = smem + OFF_QK;
    float* s_w    = smem + OFF_W;
    float* s_ctx  = smem + OFF_CTX;
    float* s_ctx2 = smem + OFF_CTX2;
    float* s_x    = smem + OFF_X;
    float* s_q    = smem + OFF_Q;
    float* s_ao   = smem + OFF_AO;
    float* s_qbk  = smem + OFF_QBK;
    float* s_red  = smem + OFF_RED;

    const float* keys_b = keys_g + (size_t)b * (L_MAX * EMB);
    const float* vals_b = vals_g + (size_t)b * (L_MAX * EMB);

    // Kick off DMA of keys chunk 0 immediately; it streams under the whole
    // prologue (q projection + Wk fold).
    if (wid == 0) tdm_load_tile64(keys_b, s_buf0);

    // ---------------- init: target row, zero WMMA pad rows ------------------
    s_x[t]       = x_g[(size_t)b * EMB + t];
    s_x[t + 256] = x_g[(size_t)b * EMB + t + 256];
    for (int i = t; i < 8 * QK_STRIDE; i += 256) s_qk[8 * QK_STRIDE + i] = 0.f;
    for (int i = t; i < 8 * W_STRIDE;  i += 256) s_w [8 * W_STRIDE  + i] = 0.f;
    __syncthreads();

    // ---------------- q = x @ Wq^T + bq -------------------------------------
    for (int rep = 0; rep < 2; ++rep) {
        int n = t + rep * 256;
        const float4* wrow = (const float4*)(Wq + (size_t)n * EMB);
        const float4* xr   = (const float4*)s_x;
        float acc = bq[n];
        #pragma unroll 4
        for (int i = 0; i < EMB / 4; ++i) {
            float4 w4 = wrow[i]; float4 x4 = xr[i];
            acc += w4.x * x4.x + w4.y * x4.y + w4.z * x4.z + w4.w * x4.w;
        }
        s_q[n] = acc;
    }
    __syncthreads();

    // ------- fold Wk into q: qk[h][k] = sum_{j<64} q[64h+j]*Wk[64h+j][k] ----
    if (t < NH) {
        float a = 0.f;
        for (int j = 0; j < HD; ++j) a += s_q[t * HD + j] * bk[t * HD + j];
        s_qbk[t] = a;
    }
    for (int rep = 0; rep < 16; ++rep) {
        int idx = rep * 256 + t;
        int h = idx >> 9;
        int k = idx & 511;
        const float* wkcol = Wk + (size_t)h * HD * EMB + k;
        const float* qh    = s_q + h * HD;
        float acc = 0.f;
        #pragma unroll 8
        for (int j = 0; j < HD; ++j) acc += qh[j] * wkcol[(size_t)j * EMB];
        s_qk[h * QK_STRIDE + k] = acc;
    }
    __syncthreads();

    // ------- scores[128 x 16] = keys @ qk^T, TDM double-buffered ------------
    const int m0 = wid * 16;
    v8f acc = {0.f, 0.f, 0.f, 0.f, 0.f, 0.f, 0.f, 0.f};
    for (int kc = 0; kc < 8; ++kc) {
        float* cur = (kc & 1) ? s_buf1 : s_buf0;
        if (wid == 0) {
            if (kc < 7) {   // queue next chunk BEFORE waiting -> DMA never idles
                tdm_load_tile64(keys_b + (kc + 1) * 64, (kc & 1) ? s_buf0 : s_buf1);
                __builtin_amdgcn_s_wait_tensorcnt(1);
            } else {
                __builtin_amdgcn_s_wait_tensorcnt(0);
            }
        }
        __syncthreads();                       // tile visible to all waves
        const float* pa = cur + (m0 + lm) * BUF_STRIDE + 2 * half;   // A: keys
        const float* pb = s_qk + lm * QK_STRIDE + kc * 64 + 2 * half;// B: qk^T
        #pragma unroll
        for (int kk = 0; kk < 16; ++kk) {
            v2f a  = *(const v2f*)(pa + 4 * kk);
            v2f bb = *(const v2f*)(pb + 4 * kk);
            acc = __builtin_amdgcn_wmma_f32_16x16x4_f32(
                      false, a, false, bb, (short)0, acc, false, false);
        }
        __syncthreads();                       // done reading before overwrite
    }
    {   // D: VGPR r -> (M = r | r+8, N = lane%16); scale = 1/(sqrt(64)*0.5)
        int h = lm;
        if (h < NH) {
            float qb = s_qbk[h];
            int lbase = m0 + half * 8;
            #pragma unroll
            for (int r = 0; r < 8; ++r)
                s_w[h * W_STRIDE + lbase + r] = (acc[r] + qb) * 0.25f;
        }
    }
    // Prefetch values chunk 0 so the DMA streams under the softmax.
    if (wid == 0) tdm_load_tile64(vals_b, s_buf0);
    __syncthreads();

    // ---------------- masked softmax per head (wave h = head h) -------------
    if (wid < NH) {
        int h = wid;
        float sv[4];
        float mx = -INFINITY;
        #pragma unroll
        for (int j = 0; j < 4; ++j) {
            int l = lane + 32 * j;
            float s = s_w[h * W_STRIDE + l];
            s = (l < len) ? s : -INFINITY;
            sv[j] = s;
            mx = fmaxf(mx, s);
        }
        #pragma unroll
        for (int d = 16; d >= 1; d >>= 1) mx = fmaxf(mx, __shfl_xor(mx, d, 32));
        float sum = 0.f;
        #pragma unroll
        for (int j = 0; j < 4; ++j) { sv[j] = __expf(sv[j] - mx); sum += sv[j]; }
        #pragma unroll
        for (int d = 16; d >= 1; d >>= 1) sum += __shfl_xor(sum, d, 32);
        float inv = 1.f / sum;
        #pragma unroll
        for (int j = 0; j < 4; ++j) {
            int l = lane + 32 * j;
            float wgt = sv[j] * inv;            // exactly 0 on padded l
            s_w[h * W_STRIDE + l] = wgt;
            out_w[(size_t)b * (NH * L_MAX) + h * L_MAX + l] = wgt;
        }
    }
    __syncthreads();

    // ------- ctx = weights[8x128] @ vals[128x512], split-K over wave pairs --
    const int tn    = wid & 3;          // N-tile within 64-col chunk
    const int khalf = wid >> 2;         // K (=L) half: 0 -> l<64, 1 -> l>=64
    const int n0c   = tn * 16;
    float* ctx_dst  = khalf ? s_ctx2 : s_ctx;
    for (int kc = 0; kc < 8; ++kc) {
        float* cur = (kc & 1) ? s_buf1 : s_buf0;
        if (wid == 0) {
            if (kc < 7) {
                tdm_load_tile64(vals_b + (kc + 1) * 64, (kc & 1) ? s_buf0 : s_buf1);
                __builtin_amdgcn_s_wait_tensorcnt(1);
            } else {
                __builtin_amdgcn_s_wait_tensorcnt(0);
            }
        }
        __syncthreads();
        v8f cc = {0.f, 0.f, 0.f, 0.f, 0.f, 0.f, 0.f, 0.f};
        const float* pa  = s_w + lm * W_STRIDE + khalf * 64 + 2 * half; // A: w
        const float* pbv = cur + n0c + lm + (khalf * 64 + 2 * half) * BUF_STRIDE;
        #pragma unroll
        for (int kk = 0; kk < 16; ++kk) {
            v2f a = *(const v2f*)(pa + 4 * kk);
            v2f bb;
            bb.x = pbv[(4 * kk)     * BUF_STRIDE];
            bb.y = pbv[(4 * kk + 1) * BUF_STRIDE];
            cc = __builtin_amdgcn_wmma_f32_16x16x4_f32(
                     false, a, false, bb, (short)0, cc, false, false);
        }
        if (lane < 16) {                // rows 0..7 are the real heads
            #pragma unroll
            for (int r = 0; r < 8; ++r)
                ctx_dst[r * CTX_STRIDE + kc * 64 + n0c + lane] = cc[r];
        }
        __syncthreads();
    }

    // ------- attn_out[e] = (ctx0+ctx1)[h(e)] . Wv[e] + bv[e] ----------------
    for (int rep = 0; rep < 2; ++rep) {
        int e = t + rep * 256;
        int h = e >> 6;
        const float4* wrow = (const float4*)(Wv + (size_t)e * EMB);
        const float4* cx   = (const float4*)(s_ctx  + h * CTX_STRIDE);
        const float4* cx2  = (const float4*)(s_ctx2 + h * CTX_STRIDE);
        float a = bv[e];
        #pragma unroll 4
        for (int i = 0; i < EMB / 4; ++i) {
            float4 w4 = wrow[i];
            float4 c4 = cx[i];
            float4 d4 = cx2[i];
            a += w4.x * (c4.x + d4.x) + w4.y * (c4.y + d4.y)
               + w4.z * (c4.z + d4.z) + w4.w * (c4.w + d4.w);
        }
        s_ao[e] = a;
    }
    __syncthreads();

    // ------- out = attn_out @ Wo^T + bo + x ; then LayerNorm ----------------
    float outv[2];
    for (int rep = 0; rep < 2; ++rep) {
        int e = t + rep * 256;
        const float4* wrow = (const float4*)(Wo + (size_t)e * EMB);
        const float4* ar   = (const float4*)s_ao;
        float a = bo[e] + s_x[e];
        #pragma unroll 4
        for (int i = 0; i < EMB / 4; ++i) {
            float4 w4 = wrow[i]; float4 a4 = ar[i];
            a += w4.x * a4.x + w4.y * a4.y + w4.z * a4.z + w4.w * a4.w;
        }
        outv[rep] = a;
    }
    float ssum = outv[0] + outv[1];
    float ssq  = outv[0] * outv[0] + outv[1] * outv[1];
    #pragma unroll
    for (int d = 16; d >= 1; d >>= 1) {
        ssum += __shfl_xor(ssum, d, 32);
        ssq  += __shfl_xor(ssq,  d, 32);
    }
    if (lane == 0) { s_red[wid] = ssum; s_red[8 + wid] = ssq; }
    __syncthreads();
    float tot = 0.f, totsq = 0.f;
    #pragma unroll
    for (int i = 0; i < 8; ++i) { tot += s_red[i]; totsq += s_red[8 + i]; }
    float mean = tot * (1.f / EMB);
    float var  = totsq * (1.f / EMB) - mean * mean;
    float inv  = rsqrtf(var + LN_EPS);
    for (int rep = 0; rep < 2; ++rep) {
        int e = t + rep * 256;
        out_norm[(size_t)b * EMB + e] = (outv[rep] - mean) * inv * gamma[e] + beta[e];
    }
}

extern "C" void kernel_launch(void* const* d_in, const int* in_sizes, int n_in,
                              void* d_out, int out_size, void* d_ws, size_t ws_size,
                              hipStream_t stream) {
    (void)in_sizes; (void)n_in; (void)out_size; (void)d_ws; (void)ws_size;
    const float* x     = (const float*)d_in[0];
    const float* keys  = (const float*)d_in[1];
    const float* vals  = (const float*)d_in[2];
    const int*   lens  = (const int*)  d_in[3];
    const float* Wq    = (const float*)d_in[4];
    const float* bq    = (const float*)d_in[5];
    const float* Wk    = (const float*)d_in[6];
    const float* bk    = (const float*)d_in[7];
    const float* Wv    = (const float*)d_in[8];
    const float* bv    = (const float*)d_in[9];
    const float* Wo    = (const float*)d_in[10];
    const float* bo    = (const float*)d_in[11];
    const float* gamma = (const float*)d_in[12];
    const float* beta  = (const float*)d_in[13];
    float* out_norm = (float*)d_out;
    float* out_w    = out_norm + (size_t)B_SZ * EMB;

    size_t smem = SMEM_FLOATS * sizeof(float);
    TargetAwareAttention_fused<<<dim3(B_SZ), dim3(256), smem, stream>>>(
        x, keys, vals, lens, Wq, bq, Wk, bk, Wv, bv, Wo, bo, gamma, beta,
        out_norm, out_w);
}